// FullTensorProduct_48052094108150
// MI455X (gfx1250) — compile-verified
//
#include <hip/hip_runtime.h>

typedef __attribute__((ext_vector_type(2))) float v2f;
typedef __attribute__((ext_vector_type(8))) float v8f;

#define ROW_IN   256      // (1+3)*64 floats per input row
#define ROW_OUT  45056    // 2*4096 + 3*12288 floats per output row
#define OFF_000  0
#define OFF_110  4096
#define OFF_011  8192
#define OFF_101  20480
#define OFF_111  32768

__device__ __forceinline__ v8f wmma4(v2f a, v2f b) {
    v8f c = {};
    // 8 args: (neg_a, A, neg_b, B, c_mod, C, reuse_a, reuse_b)
    return __builtin_amdgcn_wmma_f32_16x16x4_f32(
        false, a, false, b, (short)0, c, false, false);
}

// Build a 16x4 f32 A/B fragment for this lane from the 4 column (K) values.
// Lanes 0-15 hold K=0,1 in regs 0,1; lanes 16-31 hold K=2,3.
__device__ __forceinline__ v2f mkfrag(bool hi, float c0, float c1, float c2, float c3) {
    v2f f;
    f.x = hi ? c2 : c0;
    f.y = hi ? c3 : c1;
    return f;
}

__global__ __launch_bounds__(512)
void FullTensorProduct_48052094108150_kernel(const float* __restrict__ in1,
                                             const float* __restrict__ in2,
                                             float* __restrict__ out)
{
    const int n    = blockIdx.x;          // one row per block
    const int tid  = threadIdx.x;
    const int wave = tid >> 5;            // 16 waves per block
    const int lane = tid & 31;
    const int ut   = wave >> 2;           // u tile 0..3
    const int vt   = wave & 3;            // v tile 0..3
    const int m    = lane & 15;
    const bool hi  = (lane >= 16);

    const float* __restrict__ r1 = in1 + (size_t)n * ROW_IN;
    const float* __restrict__ r2 = in2 + (size_t)n * ROW_IN;

    const int u = ut * 16 + m;            // this lane's A row
    const int v = vt * 16 + m;            // this lane's B column

    // Per-lane feature vectors: av = (a0[u], a1[u,0..2]), bv = (b0[v], b1[v,0..2])
    const float a0u  = r1[u];
    const float a1u0 = r1[64 + 3 * u + 0];
    const float a1u1 = r1[64 + 3 * u + 1];
    const float a1u2 = r1[64 + 3 * u + 2];
    const float b0v  = r2[v];
    const float b1v0 = r2[64 + 3 * v + 0];
    const float b1v1 = r2[64 + 3 * v + 1];
    const float b1v2 = r2[64 + 3 * v + 2];

    const float s3 = 0.57735026918962576451f; // 1/sqrt(3)
    const float s2 = 0.70710678118654752440f; // 1/sqrt(2)

    // c_000 = a0 (x) b0                         (K=1 matmul)
    v8f d000 = wmma4(mkfrag(hi, a0u, 0.f, 0.f, 0.f),
                     mkfrag(hi, b0v, 0.f, 0.f, 0.f));
    // c_110 = (a1 . b1)/sqrt3                   (K=3 matmul)
    v8f d110 = wmma4(mkfrag(hi, 0.f, a1u0, a1u1, a1u2),
                     mkfrag(hi, 0.f, b1v0 * s3, b1v1 * s3, b1v2 * s3));
    // c_011[j] = a0 (x) b1_j
    v8f d011_0 = wmma4(mkfrag(hi, a0u, 0.f, 0.f, 0.f), mkfrag(hi, b1v0, 0.f, 0.f, 0.f));
    v8f d011_1 = wmma4(mkfrag(hi, a0u, 0.f, 0.f, 0.f), mkfrag(hi, b1v1, 0.f, 0.f, 0.f));
    v8f d011_2 = wmma4(mkfrag(hi, a0u, 0.f, 0.f, 0.f), mkfrag(hi, b1v2, 0.f, 0.f, 0.f));
    // c_101[i] = a1_i (x) b0
    v8f d101_0 = wmma4(mkfrag(hi, a1u0, 0.f, 0.f, 0.f), mkfrag(hi, b0v, 0.f, 0.f, 0.f));
    v8f d101_1 = wmma4(mkfrag(hi, a1u1, 0.f, 0.f, 0.f), mkfrag(hi, b0v, 0.f, 0.f, 0.f));
    v8f d101_2 = wmma4(mkfrag(hi, a1u2, 0.f, 0.f, 0.f), mkfrag(hi, b0v, 0.f, 0.f, 0.f));
    // c_111[k] = (a1_i*b1_j - a1_j*b1_i)/sqrt2, (i,j,k) cyclic   (K=2 matmul)
    v8f d111_0 = wmma4(mkfrag(hi, a1u1, a1u2, 0.f, 0.f),
                       mkfrag(hi, b1v2 * s2, -b1v1 * s2, 0.f, 0.f)); // k=0: i=1,j=2
    v8f d111_1 = wmma4(mkfrag(hi, a1u2, a1u0, 0.f, 0.f),
                       mkfrag(hi, b1v0 * s2, -b1v2 * s2, 0.f, 0.f)); // k=1: i=2,j=0
    v8f d111_2 = wmma4(mkfrag(hi, a1u0, a1u1, 0.f, 0.f),
                       mkfrag(hi, b1v1 * s2, -b1v0 * s2, 0.f, 0.f)); // k=2: i=0,j=1

    // Stores: output never re-read and >> L2, so use non-temporal stores.
    float* __restrict__ ro = out + (size_t)n * ROW_OUT;
    const int ubase = ut * 16 + (hi ? 8 : 0); // D reg r -> row M = r + 8*(lane>=16)

#pragma unroll
    for (int r = 0; r < 8; ++r) {
        const int uv = (ubase + r) * 64 + v;  // D reg r, column N = lane&15 -> v

        __builtin_nontemporal_store(d000[r], ro + OFF_000 + uv);
        __builtin_nontemporal_store(d110[r], ro + OFF_110 + uv);

        float* p011 = ro + OFF_011 + uv * 3;
        __builtin_nontemporal_store(d011_0[r], p011 + 0);
        __builtin_nontemporal_store(d011_1[r], p011 + 1);
        __builtin_nontemporal_store(d011_2[r], p011 + 2);

        float* p101 = ro + OFF_101 + uv * 3;
        __builtin_nontemporal_store(d101_0[r], p101 + 0);
        __builtin_nontemporal_store(d101_1[r], p101 + 1);
        __builtin_nontemporal_store(d101_2[r], p101 + 2);

        float* p111 = ro + OFF_111 + uv * 3;
        __builtin_nontemporal_store(d111_0[r], p111 + 0);
        __builtin_nontemporal_store(d111_1[r], p111 + 1);
        __builtin_nontemporal_store(d111_2[r], p111 + 2);
    }
}

extern "C" void kernel_launch(void* const* d_in, const int* in_sizes, int n_in,
                              void* d_out, int out_size, void* d_ws, size_t ws_size,
                              hipStream_t stream) {
    const float* in1 = (const float*)d_in[0];
    const float* in2 = (const float*)d_in[1];
    float* out = (float*)d_out;
    const int N = in_sizes[0] / ROW_IN;   // 4096 rows
    FullTensorProduct_48052094108150_kernel<<<N, 512, 0, stream>>>(in1, in2, out);
}